// RQSBijector_42013370090033
// MI455X (gfx1250) — compile-verified
//
#include <hip/hip_runtime.h>
#include <hip/hip_bf16.h>
#include <stdint.h>

#define AS3 __attribute__((address_space(3)))

#define KBINS 8
#define RMIN (-5.0f)
#define RMAX (5.0f)
#define MINBIN (0.001f)
#define MINSLOPE (0.001f)
#define MAXSLOPE (10.0f)
#define BT 32                 // b-rows per block in the spline kernel

typedef __attribute__((ext_vector_type(2))) float v2f;
typedef __attribute__((ext_vector_type(8))) float v8f;

__device__ __forceinline__ float sigf(float v) {
    return 1.0f / (1.0f + __expf(-v));
}

// CDNA5 async global->LDS copy (ASYNCcnt-tracked, bypasses VGPRs).
__device__ __forceinline__ void async_ld_f32(uint32_t lds_byte_addr, const float* g) {
    unsigned long long ga = (unsigned long long)(size_t)g;
    asm volatile("global_load_async_to_lds_b32 %0, %1, off"
                 :: "v"(lds_byte_addr), "v"(ga) : "memory");
}

__device__ __forceinline__ void wait_async0() {
    asm volatile("s_wait_asynccnt 0" ::: "memory");
}

// ---------------------------------------------------------------------------
// Kernel 1: normalize params -> SoA knot table  table[a*D + d]
//   a = 0..8  : x_knots[0..8]
//   a = 9..17 : y_knots[0..8]
//   a = 18..26: slopes [0..8]
// ---------------------------------------------------------------------------
__global__ __launch_bounds__(256) void rqs_norm_params(
    const float* __restrict__ params, float* __restrict__ table, int D) {
    int d = blockIdx.x * 256 + threadIdx.x;
    if (d >= D) return;
    const float* p = params + (size_t)d * (3 * KBINS + 1);

    float w[KBINS], h[KBINS];
    float sw = 0.f, sh = 0.f;
#pragma unroll
    for (int k = 0; k < KBINS; ++k) { w[k] = sigf(p[k]);         sw += w[k]; }
#pragma unroll
    for (int k = 0; k < KBINS; ++k) { h[k] = sigf(p[KBINS + k]); sh += h[k]; }

    const float remaining = (RMAX - RMIN) - (float)KBINS * MINBIN;
    const float fw = remaining / sw;
    const float fh = remaining / sh;

    table[0 * D + d] = RMIN;
    table[9 * D + d] = RMIN;
    float accx = RMIN, accy = RMIN;
#pragma unroll
    for (int k = 0; k < KBINS; ++k) {
        accx += MINBIN + w[k] * fw;
        accy += MINBIN + h[k] * fh;
        table[(1 + k)  * D + d] = accx;
        table[(10 + k) * D + d] = accy;
    }
#pragma unroll
    for (int j = 0; j < KBINS + 1; ++j)
        table[(18 + j) * D + d] = MINSLOPE + sigf(p[2 * KBINS + j]) * (MAXSLOPE - MINSLOPE);
}

// ---------------------------------------------------------------------------
// Kernel 2: spline evaluation. Block = 256 columns x BT rows.
// Knot slice staged to LDS via async-to-LDS; SoA stride-256 layout makes the
// bin-indexed gathers bank-conflict-free. Writes y and per-(row,colblock)
// logdet partials (no global atomics).
// ---------------------------------------------------------------------------
__global__ __launch_bounds__(256) void rqs_spline(
    const float* __restrict__ x, const float* __restrict__ table,
    float* __restrict__ y, float* __restrict__ part, int Dim) {
    __shared__ float s_tab[27 * 256];
    __shared__ float s_part[8][BT];

    const int t    = threadIdx.x;
    const int lane = t & 31;
    const int wv   = t >> 5;
    const int cb   = blockIdx.x;          // column block (256 cols)
    const int d0   = cb * 256;
    const int b0   = blockIdx.y * BT;
    const int d    = d0 + t;

    if (blockDim.x == 0u) s_tab[0] = 0.f; // never taken; keeps LDS "stored-to" for the optimizer

    // ---- stage knot table slice (27 x 256 f32) into LDS asynchronously ----
    {
        uint32_t lbase = (uint32_t)(size_t)(AS3 float*)&s_tab[0];
        const float* src = table + d0;
#pragma unroll
        for (int a = 0; a < 27; ++a) {
            int idx = a * 256 + t;
            async_ld_f32(lbase + (uint32_t)idx * 4u, src + (size_t)a * Dim + t);
        }
        wait_async0();
    }
    __syncthreads();

    // per-column constants into registers
    float xkr[8];                          // interior knots x_knots[1..7]
#pragma unroll
    for (int j = 1; j <= 7; ++j) xkr[j] = s_tab[j * 256 + t];
    const float s0  = s_tab[18 * 256 + t];
    const float sK  = s_tab[26 * 256 + t];
    const float ls0 = __logf(s0);
    const float lsK = __logf(sK);

#pragma unroll 4
    for (int r = 0; r < BT; ++r) {
        const size_t off = (size_t)(b0 + r) * (size_t)Dim + (size_t)d;
        const float xv = x[off];

        int bin = 0;
#pragma unroll
        for (int j = 1; j <= 7; ++j) bin += (xkr[j] <= xv) ? 1 : 0;

        const int idx = (bin << 8) + t;                 // bank-conflict-free gathers
        const float x_k  = s_tab[idx];
        const float x_k1 = s_tab[idx + 256];
        const float y_k  = s_tab[idx + 9  * 256];
        const float y_k1 = s_tab[idx + 10 * 256];
        const float d_k  = s_tab[idx + 18 * 256];
        const float d_k1 = s_tab[idx + 19 * 256];

        const float wid  = x_k1 - x_k;
        const float hei  = y_k1 - y_k;
        const float invw = __builtin_amdgcn_rcpf(wid);
        const float xi   = (xv - x_k) * invw;
        const float sl   = hei * invw;
        const float xi1  = 1.0f - xi;
        const float xx   = xi * xi1;
        const float num  = sl * xi * xi + d_k * xx;
        const float den  = sl + (d_k1 + d_k - 2.0f * sl) * xx;
        const float invd = __builtin_amdgcn_rcpf(den);
        const float ysp  = y_k + hei * num * invd;
        const float dnum = sl * sl * (d_k1 * xi * xi + 2.0f * sl * xx + d_k * xi1 * xi1);
        const float der  = dnum * invd * invd;

        const bool below = xv <= RMIN;
        const bool above = xv >= RMAX;
        const float yb = RMIN + (xv - RMIN) * s0;
        const float ya = RMAX + (xv - RMAX) * sK;

        y[off] = below ? yb : (above ? ya : ysp);

        float ld = below ? ls0 : (above ? lsK : __logf(der));
        // wave32 row-sum across 32 columns
#pragma unroll
        for (int m = 16; m >= 1; m >>= 1) ld += __shfl_xor(ld, m, 32);
        if (lane == 0) s_part[wv][r] = ld;
    }
    __syncthreads();

    if (t < BT) {
        float sum = 0.f;
#pragma unroll
        for (int w8 = 0; w8 < 8; ++w8) sum += s_part[w8][t];
        part[(size_t)(b0 + t) * 8 + cb] = sum;   // 8 column blocks (D = 8*256)
    }
}

// ---------------------------------------------------------------------------
// Kernel 3: reduce 8 partials/row -> logdet[row] with V_WMMA_F32_16X16X4_F32.
// A = ones(16x4)  =>  D[m][n] = sum_k B[k][n] + C[m][n]  (exact f32 row sums).
// One wave finalizes 16 rows with two chained WMMAs.
// ---------------------------------------------------------------------------
__global__ __launch_bounds__(256) void rqs_logdet_reduce(
    const float* __restrict__ part, float* __restrict__ logdet, int Brows) {
    const int lane = threadIdx.x & 31;
    const int wg   = blockIdx.x * (blockDim.x >> 5) + (threadIdx.x >> 5);
    const int row0 = wg * 16;
    if (row0 >= Brows) return;            // wave-uniform; EXEC stays all-1s for WMMA

    const int r  = lane & 15;             // B-matrix column n = b-row within group
    const int kh = lane >> 4;             // which K-slot half this lane supplies
    const float* pr = part + (size_t)(row0 + r) * 8;

    v2f a  = {1.0f, 1.0f};
    v2f b1 = { pr[kh + 0], pr[kh + 2] };  // partials c = 0..3
    v2f b2 = { pr[kh + 4], pr[kh + 6] };  // partials c = 4..7
    v8f c  = {0.f, 0.f, 0.f, 0.f, 0.f, 0.f, 0.f, 0.f};

    c = __builtin_amdgcn_wmma_f32_16x16x4_f32(false, a, false, b1, (short)0, c, false, false);
    c = __builtin_amdgcn_wmma_f32_16x16x4_f32(false, a, false, b2, (short)0, c, false, false);

    if (lane < 16) logdet[row0 + lane] = c[0];   // D[0][n], n = lane
}

// ---------------------------------------------------------------------------
extern "C" void kernel_launch(void* const* d_in, const int* in_sizes, int n_in,
                              void* d_out, int out_size, void* d_ws, size_t ws_size,
                              hipStream_t stream) {
    const float* x      = (const float*)d_in[0];
    const float* params = (const float*)d_in[1];
    const int D = in_sizes[1] / (3 * KBINS + 1);   // 2048
    const int B = in_sizes[0] / D;                  // 4096

    float* yout   = (float*)d_out;
    float* logdet = yout + (size_t)B * (size_t)D;

    float* table = (float*)d_ws;                    // 27*D f32
    float* part  = table + 27 * (size_t)D;          // B*8 f32

    rqs_norm_params<<<dim3((D + 255) / 256), 256, 0, stream>>>(params, table, D);
    rqs_spline<<<dim3(D / 256, B / BT), 256, 0, stream>>>(x, table, yout, part, D);
    rqs_logdet_reduce<<<dim3((B / 16 + 7) / 8), 256, 0, stream>>>(part, logdet, B);
}